// Encoder_46437186404757
// MI455X (gfx1250) — compile-verified
//
#include <hip/hip_runtime.h>

#define NN 50000
#define NE 800000
#define DIM 128
#define MTILES (NN / 16)   // 3125, exact
#define LDS_STRIDE 132     // padded row stride (floats) -> conflict-free column reads

typedef __attribute__((ext_vector_type(2))) float v2f;
typedef __attribute__((ext_vector_type(8))) float v8f;

// ---------------- zero scratch (agg1, agg2, cnt) ----------------
__global__ void zero_kernel(float* __restrict__ p, long long n) {
    long long i = (long long)blockIdx.x * blockDim.x + threadIdx.x;
    if (i < n) p[i] = 0.0f;
}

// ------- per-edge weight, index narrowing, degree count -------
__global__ void prep_kernel(const long long* __restrict__ ei,
                            const float* __restrict__ ew,
                            const int* __restrict__ cell_len_p,
                            float* __restrict__ w,
                            int* __restrict__ src32,
                            int* __restrict__ dst32,
                            float* __restrict__ cnt) {
    int e = blockIdx.x * blockDim.x + threadIdx.x;
    if (e >= NE) return;
    int cl = cell_len_p[0];
    int s = (int)ei[e];
    int d = (int)ei[NE + e];
    int ts = (s > cl) ? 1 : 0;
    int td = (d > cl) ? 1 : 0;
    float tw = (ts == td) ? 0.3f : 1.0f;
    w[e] = tw * ew[e];
    src32[e] = s;
    dst32[e] = d;
    unsafeAtomicAdd(&cnt[d], 1.0f);   // degree (same for both layers)
}

// ------- scatter: one wave per edge, float4 gather + fp32 atomics -------
__global__ void scatter_kernel(const float* __restrict__ xin,
                               const float* __restrict__ w,
                               const int* __restrict__ src,
                               const int* __restrict__ dst,
                               float* __restrict__ agg) {
    long long gid = (long long)blockIdx.x * blockDim.x + threadIdx.x;
    long long e = gid >> 5;           // wave32: one wave per edge
    if (e >= NE) return;
    int lane = (int)(gid & 31);
    int s = src[e];
    int d = dst[e];
    float we = w[e];
    const float4* xr = (const float4*)(xin + (long long)s * DIM);
    float4 v = xr[lane];              // lanes cover 128 floats = 32 x float4
    float* ad = agg + (long long)d * DIM + lane * 4;
    unsafeAtomicAdd(ad + 0, v.x * we);
    unsafeAtomicAdd(ad + 1, v.y * we);
    unsafeAtomicAdd(ad + 2, v.z * we);
    unsafeAtomicAdd(ad + 3, v.w * we);
}

// ------- fused: out = agg/max(cnt,1) + A @ Wr^T  (WMMA f32 16x16x4) -------
// Wr staged in LDS (padded stride); one wave computes a 16x128 output stripe.
__global__ void __launch_bounds__(128)
gemm_bias_kernel(const float* __restrict__ A,    // [NN,128]
                 const float* __restrict__ Wr,   // [128,128]
                 const float* __restrict__ agg,  // [NN,128]
                 const float* __restrict__ cnt,  // [NN]
                 float* __restrict__ out) {      // [NN,128]
    __shared__ float bsm[DIM * LDS_STRIDE];      // 67.6 KB of 320 KB/WGP

    // --- cooperative stage: all 128 threads copy one Wr row each ---
    {
        int t = threadIdx.x;                     // 0..127 == row n
        const float4* wrow = (const float4*)(Wr + (long long)t * DIM);
        float* drow = bsm + t * LDS_STRIDE;
#pragma unroll
        for (int i = 0; i < DIM / 4; i++) {
            float4 v = wrow[i];
            *(float4*)(drow + i * 4) = v;        // ds_store_b128
        }
    }
    __syncthreads();                             // all threads reach barrier

    int wave = threadIdx.x >> 5;
    int lane = threadIdx.x & 31;
    int tile = blockIdx.x * 4 + wave;            // 4 waves per block
    if (tile >= MTILES) return;                  // wave-uniform -> EXEC all-1s below
    int m0 = tile * 16;
    int rn   = lane & 15;                        // row index for A, col index for B
    int koff = (lane >> 4) * 2;                  // K sub-offset per ISA 16x4 layout

    v8f acc[8];
#pragma unroll
    for (int i = 0; i < 8; i++)
        acc[i] = (v8f){0.f, 0.f, 0.f, 0.f, 0.f, 0.f, 0.f, 0.f};

    const float* arow = A + (long long)(m0 + rn) * DIM + koff;
    const float* bl   = bsm + rn * LDS_STRIDE + koff;   // B[k][n] = Wr[n][k]

    v2f a = *(const v2f*)(arow);                 // prime A pipeline
#pragma unroll 2
    for (int k = 0; k < DIM; k += 4) {
        // batch all 8 B fragments into distinct regs (one wait, 8 MMAs)
        v2f b[8];
#pragma unroll
        for (int nt = 0; nt < 8; nt++)
            b[nt] = *(const v2f*)(bl + nt * 16 * LDS_STRIDE + k);   // ds_load_b64
        // prefetch next A fragment before the MMA burst
        v2f an = a;
        if (k + 4 < DIM) an = *(const v2f*)(arow + k + 4);
#pragma unroll
        for (int nt = 0; nt < 8; nt++)
            acc[nt] = __builtin_amdgcn_wmma_f32_16x16x4_f32(
                false, a, false, b[nt], (short)0, acc[nt], false, false);
        a = an;
    }

    // Epilogue: C/D layout -> VGPR v holds row m0+v (lanes 0-15) / m0+8+v (lanes 16-31)
    int mbase = m0 + ((lane >> 4) ? 8 : 0);
#pragma unroll
    for (int v = 0; v < 8; v++) {
        int m = mbase + v;
        float inv = 1.0f / fmaxf(cnt[m], 1.0f);
        const float* ag = agg + (long long)m * DIM;
        float* op = out + (long long)m * DIM;
#pragma unroll
        for (int nt = 0; nt < 8; nt++) {
            int n = nt * 16 + rn;
            op[n] = acc[nt][v] + ag[n] * inv;
        }
    }
}

extern "C" void kernel_launch(void* const* d_in, const int* in_sizes, int n_in,
                              void* d_out, int out_size, void* d_ws, size_t ws_size,
                              hipStream_t stream) {
    const float*     x   = (const float*)d_in[0];
    const long long* ei  = (const long long*)d_in[1];   // [2, NE] int64
    const float*     ew  = (const float*)d_in[2];
    const float*     Wr1 = (const float*)d_in[3];
    const float*     Wr2 = (const float*)d_in[4];
    const int*       cl  = (const int*)d_in[5];
    float* out = (float*)d_out;

    char* ws = (char*)d_ws;
    size_t off = 0;
    float* w     = (float*)(ws + off); off += (size_t)NE * 4;        // 3.2 MB
    int*   src32 = (int*)  (ws + off); off += (size_t)NE * 4;        // 3.2 MB
    int*   dst32 = (int*)  (ws + off); off += (size_t)NE * 4;        // 3.2 MB
    float* h     = (float*)(ws + off); off += (size_t)NN * DIM * 4;  // 25.6 MB
    // zero region: agg1 | agg2 | cnt (contiguous)
    float* agg1  = (float*)(ws + off); off += (size_t)NN * DIM * 4;  // 25.6 MB
    float* agg2  = (float*)(ws + off); off += (size_t)NN * DIM * 4;  // 25.6 MB
    float* cnt   = (float*)(ws + off); off += (size_t)NN * 4;        // 0.2 MB

    const long long nzero = (long long)NN * DIM * 2 + NN;  // agg1+agg2+cnt floats
    zero_kernel<<<(int)((nzero + 255) / 256), 256, 0, stream>>>(agg1, nzero);

    prep_kernel<<<(NE + 255) / 256, 256, 0, stream>>>(ei, ew, cl, w, src32, dst32, cnt);

    // layer 1
    const long long sthreads = (long long)NE * 32;
    scatter_kernel<<<(int)(sthreads / 256), 256, 0, stream>>>(x, w, src32, dst32, agg1);
    gemm_bias_kernel<<<(MTILES + 3) / 4, 128, 0, stream>>>(x, Wr1, agg1, cnt, h);

    // layer 2
    scatter_kernel<<<(int)(sthreads / 256), 256, 0, stream>>>(h, w, src32, dst32, agg2);
    gemm_bias_kernel<<<(MTILES + 3) / 4, 128, 0, stream>>>(h, Wr2, agg2, cnt, out);
}